// PatchGramPreprocessing_81587198755295
// MI455X (gfx1250) — compile-verified
//
#include <hip/hip_runtime.h>

typedef float v2f __attribute__((ext_vector_type(2)));
typedef float v8f __attribute__((ext_vector_type(8)));
typedef int   v4i __attribute__((__vector_size__(16)));

#define AS_GLOBAL __attribute__((address_space(1)))
#define AS_LOCAL  __attribute__((address_space(3)))

typedef AS_GLOBAL v4i* gv4i_p;   // global (AS1) int4 pointer
typedef AS_LOCAL  v4i* lv4i_p;   // LDS (AS3) int4 pointer

#if __has_builtin(__builtin_amdgcn_global_load_async_to_lds_b128)
#define USE_ASYNC_LDS 1
#else
#define USE_ASYNC_LDS 0
#endif

static __device__ __forceinline__ void wait_asynccnt0() {
#if __has_builtin(__builtin_amdgcn_s_wait_asynccnt)
    __builtin_amdgcn_s_wait_asynccnt(0);
#else
    asm volatile("s_wait_asynccnt 0x0" ::: "memory");
#endif
}

// One block = one (sample, layer). 128 threads = 4 waves (wave32).
// Wave w computes gram rows [16w, 16w+16) x all 64 cols as 4 WMMA tiles.
template <int C, int KPOOL>
__global__ __launch_bounds__(128) void patch_gram_kernel(
    const float* __restrict__ feat, float* __restrict__ out, int layer, int N)
{
    constexpr int P = 9;
    constexpr int CP = C * P;           // floats of input per sample
    constexpr int GS = 12;              // g row stride (K padded 9 -> 12)
    constexpr int GR = 68;              // gram row stride (64 + 4 pad, conflict-free)

    __shared__ float raw[CP];           // staged input  (18KB / 36KB)
    __shared__ float g[64 * GS];        // pooled 64 x 12  (3KB)
    __shared__ float gram[64 * GR];     // 64 x 64 gram, padded (17KB)

    const int n   = blockIdx.x;
    const int tid = threadIdx.x;
    if (n >= N) return;

    // ---- 1. Stage raw sample into LDS (CP % 512 == 0 -> uniform loops) ----
#if USE_ASYNC_LDS
    {
        // ASYNCcnt-tracked DMA: 16B per lane per issue, no VGPR staging.
        const char* gsrc = (const char*)(feat + (size_t)n * CP);
        #pragma unroll 4
        for (int i = tid; i < CP / 4; i += 128) {
            __builtin_amdgcn_global_load_async_to_lds_b128(
                (gv4i_p)(gsrc + 16 * (size_t)i),
                (lv4i_p)((char*)raw + 16 * i),
                /*offset=*/0, /*cpol=*/0);
        }
    }
#else
    {
        const float4* src = reinterpret_cast<const float4*>(feat + (size_t)n * CP);
        float4* dst = reinterpret_cast<float4*>(raw);
        #pragma unroll 4
        for (int i = tid; i < CP / 4; i += 128) dst[i] = src[i];
    }
#endif
    // zero-fill g (covers the K=9..11 padding); 768 = 6*128, uniform
    for (int i = tid; i < 64 * GS; i += 128) g[i] = 0.0f;
#if USE_ASYNC_LDS
    wait_asynccnt0();                   // this wave's DMA landed in LDS
#endif
    __syncthreads();                    // all waves' contributions visible

    // ---- 2. Channel pooling C -> 64 (group mean of KPOOL channels) ----
    {
        constexpr float inv_k = 1.0f / (float)KPOOL;
        #pragma unroll
        for (int it = 0; it < 5; ++it) {            // ceil(576/128)
            int idx = tid + it * 128;
            if (idx < 64 * P) {
                int c = idx / P, p = idx - c * P;
                float s = 0.0f;
                #pragma unroll
                for (int j = 0; j < KPOOL; ++j) s += raw[(c * KPOOL + j) * P + p];
                g[c * GS + p] = s * inv_k;
            }
        }
    }
    __syncthreads();

    // ---- 3. Gram = G * G^T via V_WMMA_F32_16X16X4_F32 ----
    // A 16x4 f32 layout (ISA 7.12.2): lane m=lane&15 is row M; lanes<16 hold
    // K={ks,ks+1} in {v0,v1}, lanes>=16 hold K={ks+2,ks+3}. B (4x16) is the
    // symmetric gather from g with column base of the tile.
    const int w    = tid >> 5;          // wave id: row strip
    const int lane = tid & 31;
    const int hh   = lane >> 4;         // lane half
    const int m    = lane & 15;

    v8f acc[4] = {};                    // 4 tiles of 16x16 f32
    #pragma unroll
    for (int ks = 0; ks < 12; ks += 4) {
        v2f a;
        const int arow = (16 * w + m) * GS + ks + 2 * hh;
        a.x = g[arow];
        a.y = g[arow + 1];
        #pragma unroll
        for (int t = 0; t < 4; ++t) {
            v2f b;
            const int brow = (16 * t + m) * GS + ks + 2 * hh;
            b.x = g[brow];
            b.y = g[brow + 1];
            acc[t] = __builtin_amdgcn_wmma_f32_16x16x4_f32(
                /*neg_a=*/false, a, /*neg_b=*/false, b,
                /*c_mod=*/(short)0, acc[t], /*reuse_a=*/false, /*reuse_b=*/false);
        }
    }

    // ---- 4. Spill gram strips to LDS (C/D layout: VGPR r holds M=r / M=r+8) --
    #pragma unroll
    for (int t = 0; t < 4; ++t) {
        #pragma unroll
        for (int r = 0; r < 8; ++r) {
            gram[(16 * w + r + 8 * hh) * GR + 16 * t + m] = acc[t][r];
        }
    }
    __syncthreads();

    // ---- 5. Final pooling: flat 4096 -> 1024 (4 consecutive cols per row),
    //         fold in /P (=9) and /4: scale = 1/36. Coalesced NT f32 stores. --
    {
        constexpr float sc = 1.0f / 36.0f;
        float* outp = out + (size_t)n * 2048 + (size_t)layer * 1024;
        #pragma unroll
        for (int it = 0; it < 8; ++it) {            // 1024 = 8*128, uniform
            int o  = tid + it * 128;
            int c  = o >> 4;
            int dg = o & 15;
            int base = c * GR + 4 * dg;
            float v = (gram[base] + gram[base + 1] + gram[base + 2] + gram[base + 3]) * sc;
            __builtin_nontemporal_store(v, &outp[o]);
        }
    }
}

extern "C" void kernel_launch(void* const* d_in, const int* in_sizes, int n_in,
                              void* d_out, int out_size, void* d_ws, size_t ws_size,
                              hipStream_t stream)
{
    const float* feat0 = (const float*)d_in[0];   // [N, 512, 3, 3]
    const float* feat1 = (const float*)d_in[1];   // [N, 1024, 3, 3]
    float* out = (float*)d_out;                   // [N, 2, 1024]

    const int N0 = in_sizes[0] / (512 * 9);
    const int N1 = in_sizes[1] / (1024 * 9);

    patch_gram_kernel<512, 8><<<dim3(N0), dim3(128), 0, stream>>>(feat0, out, 0, N0);
    patch_gram_kernel<1024, 16><<<dim3(N1), dim3(128), 0, stream>>>(feat1, out, 1, N1);
}